// DecoderAttentionRNN_8315056685069
// MI455X (gfx1250) — compile-verified
//
#include <hip/hip_runtime.h>
#include <hip/hip_bf16.h>

// ---- model dims (hardwired, match reference) ----
#define L_   2
#define T_   128
#define NB   64      // batch N
#define H_   512
#define E_   256
#define A_   200
#define AP   208     // A padded to 13*16 for WMMA tiles
#define V_   1000
#define S_   40
#define H3   1536
#define SOS_ 1

#define MT 4         // 4 x 16 = 64 rows per wave (covers full batch)
#define NT 2         // 2 x 16 = 32 cols per wave

typedef float v2f __attribute__((ext_vector_type(2)));
typedef float v8f __attribute__((ext_vector_type(8)));

// =====================================================================
// fp32 WMMA GEMM:  C[z] = A[z] (MxK, row-major, lda) * W[z]^T + bias[z]
// W is (Nout, K) row-major (PyTorch Linear weight).
// One wave (32 thr) per 64(M) x 32(N) strip: 8 accumulators; the B (weight)
// fragment is loaded once per k-step and reused across 4 row tiles, so the
// big weight matrix streams through L2 exactly once per GEMM.
// Ragged column tiles (Nout=200/1000) take a separate masked path so the
// hot loop has no mask multiplies.
// =====================================================================
__global__ void gemm_wmma_f32(const float* __restrict__ A, long sAz, int lda,
                              const float* __restrict__ W, long sWz,
                              const float* __restrict__ bias, long sBz,
                              float* __restrict__ C, long sCz, int ldc,
                              int K, int NoutReal, int relu)
{
    const int z = blockIdx.z;
    const float* Az = A + (size_t)z * sAz;
    const float* Wz = W + (size_t)z * sWz;
    const float* bz = bias + (size_t)z * sBz;
    float* Cz = C + (size_t)z * sCz;

    const int lane = threadIdx.x;          // 0..31, wave32
    const int m    = lane & 15;            // row within tile (A) / col (B,C)
    const int half = lane >> 4;            // selects K pair {0,1} vs {2,3}
    const int m0   = blockIdx.x * (16 * MT);
    const int n0   = blockIdx.y * (16 * NT);

    // ISA 7.12.2: A 16x4 fp32 -> lane m holds K = 2*half, 2*half+1 in 2 VGPRs
    const float* arow = Az + (size_t)(m0 + m) * lda + 2 * half;
    const size_t astep = (size_t)16 * lda;

    int          nc[NT];
    float        wm[NT];
    const float* wrow[NT];
#pragma unroll
    for (int j = 0; j < NT; ++j) {
        nc[j] = n0 + 16 * j + m;
        wm[j] = (nc[j] < NoutReal) ? 1.0f : 0.0f;
        int ncc = (nc[j] < NoutReal) ? nc[j] : (NoutReal - 1);
        wrow[j] = Wz + (size_t)ncc * K + 2 * half;   // B 4x16: lane n holds K pair
    }

    v8f acc[MT][NT];
#pragma unroll
    for (int i = 0; i < MT; ++i)
#pragma unroll
        for (int j = 0; j < NT; ++j)
            acc[i][j] = (v8f){0.f, 0.f, 0.f, 0.f, 0.f, 0.f, 0.f, 0.f};

    if (n0 + 16 * NT <= NoutReal) {
        // -------- fast path: all column tiles in range, no masking --------
        for (int k = 0; k < K; k += 4) {
            v2f b[NT];
#pragma unroll
            for (int j = 0; j < NT; ++j) { b[j].x = wrow[j][k]; b[j].y = wrow[j][k + 1]; }
            v2f a[MT];
#pragma unroll
            for (int i = 0; i < MT; ++i) {
                a[i].x = arow[(size_t)i * astep + k];
                a[i].y = arow[(size_t)i * astep + k + 1];
            }
#pragma unroll
            for (int i = 0; i < MT; ++i)
#pragma unroll
                for (int j = 0; j < NT; ++j)
                    acc[i][j] = __builtin_amdgcn_wmma_f32_16x16x4_f32(
                        false, a[i], false, b[j], (short)0, acc[i][j], false, false);
        }
    } else {
        // -------- masked path: ragged trailing column tiles --------
        for (int k = 0; k < K; k += 4) {
            v2f b[NT];
#pragma unroll
            for (int j = 0; j < NT; ++j) {
                b[j].x = wrow[j][k] * wm[j];
                b[j].y = wrow[j][k + 1] * wm[j];
            }
            v2f a[MT];
#pragma unroll
            for (int i = 0; i < MT; ++i) {
                a[i].x = arow[(size_t)i * astep + k];
                a[i].y = arow[(size_t)i * astep + k + 1];
            }
#pragma unroll
            for (int i = 0; i < MT; ++i)
#pragma unroll
                for (int j = 0; j < NT; ++j)
                    acc[i][j] = __builtin_amdgcn_wmma_f32_16x16x4_f32(
                        false, a[i], false, b[j], (short)0, acc[i][j], false, false);
        }
    }

#pragma unroll
    for (int j = 0; j < NT; ++j) {
        if (nc[j] < NoutReal) {
            const float bv = bz[nc[j]];
#pragma unroll
            for (int i = 0; i < MT; ++i) {
#pragma unroll
                for (int r = 0; r < 8; ++r) {
                    // C layout: VGPR r -> row (r + 8*half), col = lane&15
                    float v = acc[i][j][r] + bv;
                    if (relu) v = fmaxf(v, 0.f);
                    Cz[(size_t)(m0 + 16 * i + r + 8 * half) * ldc + nc[j]] = v;
                }
            }
        }
    }
}

// =====================================================================
// setup kernels
// =====================================================================
__global__ void copy_f32(const float* __restrict__ src, float* __restrict__ dst, int n)
{
    int i = blockIdx.x * blockDim.x + threadIdx.x;
    if (i < n) dst[i] = src[i];
}

// embAll[s][n][e] = relu(embW[tok(s,n)][e]);  tok = s==0 ? SOS : targets[n][s-1]
__global__ void embed_all_kernel(const int* __restrict__ targets,
                                 const float* __restrict__ embW,
                                 float* __restrict__ embAll)
{
    int idx = blockIdx.x * blockDim.x + threadIdx.x;   // < S*N*E
    int e  = idx & (E_ - 1);
    int sn = idx >> 8;           // E_ = 256
    int n  = sn & (NB - 1);
    int s  = sn >> 6;            // NB = 64
    int tok = (s == 0) ? SOS_ : targets[n * S_ + (s - 1)];
    float v = embW[(size_t)tok * E_ + e];
    embAll[idx] = fmaxf(v, 0.f);
}

// =====================================================================
// attention: scores[lt][n] = Vb[l] + sum_a Vw[l][a]*tanh(qproj[l][n][a]+Kproj[lt][n][a])
// one wave per score; lane-split over A, wave32 shfl_xor reduction
// =====================================================================
__global__ void scores_kernel(const float* __restrict__ qproj,
                              const float* __restrict__ kproj,
                              const float* __restrict__ Vw,
                              const float* __restrict__ Vb,
                              float* __restrict__ scores)
{
    int wid  = (blockIdx.x * blockDim.x + threadIdx.x) >> 5;   // < L*T*N
    int lane = threadIdx.x & 31;
    int n  = wid & (NB - 1);
    int lt = wid >> 6;
    int l  = lt >> 7;                      // T_ = 128
    const float* kp = kproj + (size_t)(lt * NB + n) * AP;
    const float* qp = qproj + (size_t)(l * NB + n) * AP;
    const float* vw = Vw + l * A_;
    float acc = 0.f;
    for (int a = lane; a < A_; a += 32)
        acc += vw[a] * tanhf(qp[a] + kp[a]);
    for (int off = 16; off; off >>= 1)
        acc += __shfl_xor(acc, off, 32);
    if (lane == 0) scores[lt * NB + n] = acc + Vb[l];
}

// =====================================================================
// joint softmax over L*T (=256, one score per thread) per batch column n,
// then context[n][h] = sum_lt w*enc ; writes x = [emb | context] (N x 768)
// =====================================================================
__global__ void softmax_context_kernel(const float* __restrict__ scores,
                                       const float* __restrict__ enc,
                                       const float* __restrict__ embS,
                                       float* __restrict__ x)
{
    __shared__ float red[256];
    __shared__ float w[256];
    const int n   = blockIdx.x;
    const int tid = threadIdx.x;

    float s = scores[tid * NB + n];
    red[tid] = s; __syncthreads();
    for (int off = 128; off; off >>= 1) {
        if (tid < off) red[tid] = fmaxf(red[tid], red[tid + off]);
        __syncthreads();
    }
    float mx = red[0]; __syncthreads();
    float e = expf(s - mx);
    red[tid] = e; __syncthreads();
    for (int off = 128; off; off >>= 1) {
        if (tid < off) red[tid] += red[tid + off];
        __syncthreads();
    }
    float inv = 1.0f / red[0];
    w[tid] = e * inv; __syncthreads();

    // embedding half of x
    x[(size_t)n * (E_ + H_) + tid] = embS[(size_t)n * E_ + tid];

    // context half of x (H=512, 2 elems per thread)
    for (int h = tid; h < H_; h += 256) {
        float acc = 0.f;
#pragma unroll 4
        for (int lt = 0; lt < L_ * T_; ++lt)
            acc += w[lt] * enc[(size_t)(lt * NB + n) * H_ + h];
        x[(size_t)n * (E_ + H_) + E_ + h] = acc;
    }
}

// =====================================================================
// GRU elementwise combine (PyTorch gate order r,z,n); h updated in place
// =====================================================================
__global__ void gru_combine_kernel(const float* __restrict__ gi,
                                   const float* __restrict__ gh,
                                   float* __restrict__ h)
{
    int idx = blockIdx.x * blockDim.x + threadIdx.x;   // < N*H
    int n  = idx >> 9;           // H_ = 512
    int hh = idx & (H_ - 1);
    const float* gin = gi + (size_t)n * H3;
    const float* ghn = gh + (size_t)n * H3;
    float ir = gin[hh], iz = gin[H_ + hh], in_ = gin[2 * H_ + hh];
    float hr = ghn[hh], hz = ghn[H_ + hh], hn  = ghn[2 * H_ + hh];
    float r  = 1.f / (1.f + expf(-(ir + hr)));
    float z  = 1.f / (1.f + expf(-(iz + hz)));
    float nn = tanhf(in_ + r * hn);
    float hp = h[idx];
    h[idx] = (1.f - z) * nn + z * hp;
}

// =====================================================================
extern "C" void kernel_launch(void* const* d_in, const int* in_sizes, int n_in,
                              void* d_out, int out_size, void* d_ws, size_t ws_size,
                              hipStream_t stream)
{
    const float* enc   = (const float*)d_in[0];   // (L,T,N,H)
    const float* efs   = (const float*)d_in[1];   // (L,N,H)
    const int*   tgt   = (const int*)  d_in[2];   // (N,S)
    const float* embW  = (const float*)d_in[3];   // (V,E)
    const float* Qw    = (const float*)d_in[4];
    const float* Qb    = (const float*)d_in[5];
    const float* Kw    = (const float*)d_in[6];
    const float* Kb    = (const float*)d_in[7];
    const float* Vw    = (const float*)d_in[8];
    const float* Vb    = (const float*)d_in[9];
    const float* Wih0  = (const float*)d_in[10];
    const float* Whh0  = (const float*)d_in[11];
    const float* bih0  = (const float*)d_in[12];
    const float* bhh0  = (const float*)d_in[13];
    const float* Wih1  = (const float*)d_in[14];
    const float* Whh1  = (const float*)d_in[15];
    const float* bih1  = (const float*)d_in[16];
    const float* bhh1  = (const float*)d_in[17];
    const float* Pw0   = (const float*)d_in[18];
    const float* Pb0   = (const float*)d_in[19];
    const float* Pw1   = (const float*)d_in[20];
    const float* Pb1   = (const float*)d_in[21];
    float* out = (float*)d_out;                   // (N,S,V)

    // ---- workspace carve-up (floats) ----
    float* ws = (float*)d_ws;
    float* kproj   = ws;  ws += (size_t)L_ * T_ * NB * AP;   // 3,407,872
    float* embAll  = ws;  ws += (size_t)S_ * NB * E_;        //   655,360
    float* states  = ws;  ws += (size_t)L_ * NB * H_;        //    65,536
    float* qproj   = ws;  ws += (size_t)L_ * NB * AP;        //    26,624
    float* scoresB = ws;  ws += (size_t)L_ * T_ * NB;        //    16,384
    float* xbuf    = ws;  ws += (size_t)NB * (E_ + H_);      //    49,152
    float* gi      = ws;  ws += (size_t)NB * H3;             //    98,304
    float* gh      = ws;  ws += (size_t)NB * H3;             //    98,304
    float* projb   = ws;  ws += (size_t)NB * H_;             //    32,768

    // column-tile grid extents (NT=2 tiles of 16 per block)
    const int gyA  = (AP / 16 + NT - 1) / NT;                 // 13 -> 7
    const int gyG  = (H3 / 16) / NT;                          // 96 -> 48
    const int gyP  = (H_ / 16) / NT;                          // 32 -> 16
    const int gyV  = ((V_ + 15) / 16 + NT - 1) / NT;          // 63 -> 32

    // ---- setup (step-invariant) ----
    copy_f32<<<(L_ * NB * H_) / 256, 256, 0, stream>>>(efs, states, L_ * NB * H_);
    embed_all_kernel<<<(S_ * NB * E_) / 256, 256, 0, stream>>>(tgt, embW, embAll);
    // Kproj[l,t,n,:] = enc[l,t,n,:] @ Kw[l]^T + Kb[l]   (M=8192/layer, K=512, Nout=200)
    gemm_wmma_f32<<<dim3(T_ * NB / (16 * MT), gyA, L_), 32, 0, stream>>>(
        enc, (long)T_ * NB * H_, H_, Kw, (long)A_ * H_, Kb, A_,
        kproj, (long)T_ * NB * AP, AP, H_, A_, 0);

    // ---- sequential decode (teacher forcing) ----
    for (int s = 0; s < S_; ++s) {
        // qproj[l] = states[l] @ Qw[l]^T + Qb[l]
        gemm_wmma_f32<<<dim3(1, gyA, L_), 32, 0, stream>>>(
            states, (long)NB * H_, H_, Qw, (long)A_ * H_, Qb, A_,
            qproj, (long)NB * AP, AP, H_, A_, 0);
        // attention energies
        scores_kernel<<<(L_ * T_ * NB) / 8, 256, 0, stream>>>(qproj, kproj, Vw, Vb, scoresB);
        // softmax over L*T, context, assemble x = [emb|ctx]
        softmax_context_kernel<<<NB, 256, 0, stream>>>(
            scoresB, enc, embAll + (size_t)s * NB * E_, xbuf);

        // GRU layer 0
        gemm_wmma_f32<<<dim3(1, gyG, 1), 32, 0, stream>>>(
            xbuf, 0, E_ + H_, Wih0, 0, bih0, 0, gi, 0, H3, E_ + H_, H3, 0);
        gemm_wmma_f32<<<dim3(1, gyG, 1), 32, 0, stream>>>(
            states, 0, H_, Whh0, 0, bhh0, 0, gh, 0, H3, H_, H3, 0);
        gru_combine_kernel<<<(NB * H_) / 256, 256, 0, stream>>>(gi, gh, states);

        // GRU layer 1 (input = new h0)
        gemm_wmma_f32<<<dim3(1, gyG, 1), 32, 0, stream>>>(
            states, 0, H_, Wih1, 0, bih1, 0, gi, 0, H3, H_, H3, 0);
        gemm_wmma_f32<<<dim3(1, gyG, 1), 32, 0, stream>>>(
            states + (size_t)NB * H_, 0, H_, Whh1, 0, bhh1, 0, gh, 0, H3, H_, H3, 0);
        gru_combine_kernel<<<(NB * H_) / 256, 256, 0, stream>>>(gi, gh, states + (size_t)NB * H_);

        // projection head: relu(h1 @ Pw0^T + Pb0) @ Pw1^T + Pb1 -> out[:, s, :]
        gemm_wmma_f32<<<dim3(1, gyP, 1), 32, 0, stream>>>(
            states + (size_t)NB * H_, 0, H_, Pw0, 0, Pb0, 0, projb, 0, H_, H_, H_, 1);
        gemm_wmma_f32<<<dim3(1, gyV, 1), 32, 0, stream>>>(
            projb, 0, H_, Pw1, 0, Pb1, 0, out + (size_t)s * V_, 0, S_ * V_, H_, V_, 0);
    }
}